// VectorQuantizer_14929306320975
// MI455X (gfx1250) — compile-verified
//
#include <hip/hip_runtime.h>

// VQ-VAE vector quantizer for gfx1250 (MI455X).
// Scores via V_WMMA_F32_16X16X4_F32 (exact fp32, matches reference argmin),
// codebook staged once per WG in 260KB LDS (WGP has 320KB), 4 strips/wave to
// amortize staging, dual wmma accumulator chains for ILP, deterministic loss.

typedef float v2f __attribute__((ext_vector_type(2)));
typedef float v8f __attribute__((ext_vector_type(8)));

#define KCB 1024          // codebook entries
#define DIM 64            // embedding dim
#define WAVES_PER_WG 8
#define TPB (WAVES_PER_WG * 32)
#define NROWS 131072      // 8*64*16*16 z rows
#define NSTRIPS (NROWS / 16)                  // 8192 strips of 16 rows
#define STRIPS_PER_WAVE 4
#define NBLOCKS (NSTRIPS / (WAVES_PER_WG * STRIPS_PER_WAVE)) // 256

__global__ __launch_bounds__(TPB)
void vq_main(const float* __restrict__ z, const float* __restrict__ emb,
             float* __restrict__ qout, float* __restrict__ idxout,
             float* __restrict__ partials)
{
    extern __shared__ float smem[];          // KCB*DIM codebook + KCB norms
    float* s_emb = smem;
    float* s_nrm = smem + KCB * DIM;

    const int tid = threadIdx.x;

    // ---- stage codebook into LDS (16B vector loads, once per workgroup) ----
    {
        const float4* src = (const float4*)emb;
        float4* dst = (float4*)s_emb;
        for (int i = tid; i < (KCB * DIM) / 4; i += TPB) dst[i] = src[i];
    }
    __syncthreads();
    // per-entry squared norms
    for (int k = tid; k < KCB; k += TPB) {
        float s = 0.f;
        #pragma unroll 8
        for (int d = 0; d < DIM; ++d) { float v = s_emb[k * DIM + d]; s += v * v; }
        s_nrm[k] = s;
    }
    __syncthreads();

    const int lane = tid & 31;
    const int wave = tid >> 5;
    const int nrow = lane & 15;                 // M index for A, N index for B/C
    const int koff = (lane < 16) ? 0 : 2;       // K sub-offset per 16x4 fragment

    for (int sp = 0; sp < STRIPS_PER_WAVE; ++sp) {
        const int strip = (blockIdx.x * WAVES_PER_WG + wave) * STRIPS_PER_WAVE + sp;
        const int rowBase = strip * 16;

        // ---- A fragments: 16 rows x 64, as 16 k-steps of 16x4 fp32 ----
        v2f a[16];
        #pragma unroll
        for (int s = 0; s < 16; ++s)
            a[s] = *(const v2f*)(z + (size_t)(rowBase + nrow) * DIM + 4 * s + koff);

        float bestv[8];
        int   besti[8];
        #pragma unroll
        for (int j = 0; j < 8; ++j) { bestv[j] = 3.4e38f; besti[j] = 0; }

        // ---- sweep 64 column tiles of the codebook ----
        for (int ct = 0; ct < KCB / 16; ++ct) {
            v8f acc0 = {};          // k = 0..31   (independent wmma chain)
            v8f acc1 = {};          // k = 32..63  (independent wmma chain)
            const float* bbase = s_emb + (size_t)(ct * 16 + nrow) * DIM + koff;
            #pragma unroll
            for (int s = 0; s < 8; ++s) {
                v2f b0 = *(const v2f*)(bbase + 4 * s);
                v2f b1 = *(const v2f*)(bbase + 4 * (s + 8));
                acc0 = __builtin_amdgcn_wmma_f32_16x16x4_f32(
                    false, a[s],     false, b0, (short)0, acc0, false, false);
                acc1 = __builtin_amdgcn_wmma_f32_16x16x4_f32(
                    false, a[s + 8], false, b1, (short)0, acc1, false, false);
            }
            const float en = s_nrm[ct * 16 + nrow];
            const int   k  = ct * 16 + nrow;
            #pragma unroll
            for (int j = 0; j < 8; ++j) {
                // ||e_k||^2 - 2 z.e  (row-constant ||z||^2 dropped for argmin)
                float score = en - 2.0f * (acc0[j] + acc1[j]);
                if (score < bestv[j]) { bestv[j] = score; besti[j] = k; }
            }
        }

        // ---- butterfly min-reduce across the 16-lane N groups ----
        #pragma unroll
        for (int j = 0; j < 8; ++j) {
            #pragma unroll
            for (int m = 1; m <= 8; m <<= 1) {
                float ov = __shfl_xor(bestv[j], m, 32);
                int   oi = __shfl_xor(besti[j], m, 32);
                if (ov < bestv[j] || (ov == bestv[j] && oi < besti[j])) {
                    bestv[j] = ov; besti[j] = oi;
                }
            }
            // lanes 0..15 hold row j result; lanes 16..31 hold row j+8
            if (nrow == 0)
                idxout[rowBase + j + (lane >> 4) * 8] = (float)besti[j];
        }

        // ---- gather winning codebook rows, write quantized, accum loss ----
        float lsum = 0.f;
        #pragma unroll
        for (int r = 0; r < 16; ++r) {
            int k = __shfl(besti[r & 7], (r < 8) ? 0 : 16, 32);
            float2 e2 = *(const float2*)(s_emb + (size_t)k * DIM + lane * 2);
            size_t off = (size_t)(rowBase + r) * DIM + lane * 2;
            float2 z2 = *(const float2*)(z + off);
            *(float2*)(qout + off) = e2;
            float dx = e2.x - z2.x, dy = e2.y - z2.y;
            lsum += dx * dx + dy * dy;
        }
        #pragma unroll
        for (int m = 1; m <= 16; m <<= 1) lsum += __shfl_xor(lsum, m, 32);
        if (lane == 0) partials[strip] = lsum;   // deterministic: no atomics
    }
}

// Fixed-order reduction of NSTRIPS partials -> vq_loss = 1.5 * sum / numel
__global__ __launch_bounds__(256)
void vq_finalize(const float* __restrict__ partials, float* __restrict__ loss)
{
    __shared__ float red[256];
    const int t = threadIdx.x;
    float s = 0.f;
    for (int i = t; i < NSTRIPS; i += 256) s += partials[i];   // fixed order
    red[t] = s;
    __syncthreads();
    if (t == 0) {
        float total = 0.f;
        for (int i = 0; i < 256; ++i) total += red[i];         // fixed order
        loss[0] = 1.5f * total / (float)(NROWS * DIM);
    }
}

extern "C" void kernel_launch(void* const* d_in, const int* in_sizes, int n_in,
                              void* d_out, int out_size, void* d_ws, size_t ws_size,
                              hipStream_t stream)
{
    const float* z   = (const float*)d_in[0];   // [131072, 64] flat
    const float* emb = (const float*)d_in[1];   // [1024, 64]
    const int nz = in_sizes[0];                 // 8388608

    float* qout   = (float*)d_out;              // quantized_ste (== quantized)
    float* loss   = (float*)d_out + nz;         // scalar vq_loss
    float* idxout = (float*)d_out + nz + 1;     // idx as float
    float* partials = (float*)d_ws;             // NSTRIPS per-strip loss partials

    const size_t shmem = (size_t)(KCB * DIM + KCB) * sizeof(float); // 260KB < 320KB WGP LDS
    (void)hipFuncSetAttribute((const void*)vq_main,
                              hipFuncAttributeMaxDynamicSharedMemorySize,
                              (int)shmem);

    vq_main<<<NBLOCKS, TPB, shmem, stream>>>(z, emb, qout, idxout, partials);
    vq_finalize<<<1, 256, 0, stream>>>(partials, loss);
    (void)n_in; (void)out_size; (void)ws_size;
}